// GCN_85005992722928
// MI455X (gfx1250) — compile-verified
//
#include <hip/hip_runtime.h>
#include <hip/hip_bf16.h>

typedef __attribute__((ext_vector_type(2))) float v2f;
typedef __attribute__((ext_vector_type(8))) float v8f;

#define DFEAT 128
#define DOUT  40
#define COLP  48    // Wout cols padded to 3 WMMA tiles
#define LDSA  132   // padded LDS row stride (floats) -> conflict-free A reads

// ---------------- degree / normalization ----------------
__global__ __launch_bounds__(256) void deg_init_kernel(float* __restrict__ deg, int n) {
    int i = blockIdx.x * 256 + threadIdx.x;
    if (i < n) deg[i] = 1.0f;   // self loop counts once
}

__global__ __launch_bounds__(256) void deg_edges_kernel(const long long* __restrict__ dst,
                                                        float* __restrict__ deg, long long E) {
    long long t = (long long)blockIdx.x * 256 + threadIdx.x;
    if (t < E) atomicAdd(&deg[(int)dst[t]], 1.0f);
}

__global__ __launch_bounds__(256) void dis_finish_kernel(float* __restrict__ deg, int n) {
    int i = blockIdx.x * 256 + threadIdx.x;
    if (i < n) { float d = deg[i]; deg[i] = d > 0.f ? rsqrtf(d) : 0.f; }
}

// ---------------- weight packing for WMMA B-operand ----------------
// B fragment per lane needs W[k][col], W[k+1][col] (k even) adjacent -> enables b64 loads.
// Wp[(k>>1)*2*ncols + col*2 + (k&1)] = W[k*ncols_src + col]
__global__ __launch_bounds__(256) void pack_w128_kernel(const float* __restrict__ W,
                                                        float* __restrict__ Wp) {
    int t = blockIdx.x * 256 + threadIdx.x;     // t = k*128 + col
    if (t >= DFEAT * DFEAT) return;
    int k = t >> 7, col = t & 127;
    Wp[(size_t)(k >> 1) * (2 * DFEAT) + col * 2 + (k & 1)] = W[t];
}

__global__ __launch_bounds__(256) void pack_wout_kernel(const float* __restrict__ W,
                                                        float* __restrict__ Wp) {
    int t = blockIdx.x * 256 + threadIdx.x;     // t = k*48 + col
    if (t >= DFEAT * COLP) return;
    int k = t / COLP, col = t % COLP;
    float v = (col < DOUT) ? W[(size_t)k * DOUT + col] : 0.f;
    Wp[(size_t)(k >> 1) * (2 * COLP) + col * 2 + (k & 1)] = v;
}

// ---------------- dense GEMM: Y[n x 128] = (relu?)X[n x 128] @ W[128 x 128] ----------------
// 1 block = 16 rows, 8 waves, wave w -> 16-col tile. K=128 via 32x V_WMMA_F32_16X16X4_F32.
__global__ __launch_bounds__(256) void gemm128_kernel(const float* __restrict__ X,
                                                      const float2* __restrict__ Wp,
                                                      float* __restrict__ Y,
                                                      int n, int relu_in) {
    __shared__ float Alds[16 * LDSA];
    const int tid = threadIdx.x;
    const int rowbase = blockIdx.x * 16;
    const bool full = (rowbase + 16 <= n);

    // Stage A tile (16 x 128) into LDS; 512 float4 loads over 256 threads.
    for (int idx = tid; idx < 512; idx += 256) {
        int r = idx >> 5, c4 = idx & 31;
        int row = rowbase + r; if (row >= n) row = n - 1;
        float4 v = ((const float4*)(X + (size_t)row * DFEAT))[c4];
        if (relu_in) {
            v.x = fmaxf(v.x, 0.f); v.y = fmaxf(v.y, 0.f);
            v.z = fmaxf(v.z, 0.f); v.w = fmaxf(v.w, 0.f);
        }
        float* dp = &Alds[r * LDSA + c4 * 4];
        dp[0] = v.x; dp[1] = v.y; dp[2] = v.z; dp[3] = v.w;
    }
    __syncthreads();

    const int lane = tid & 31;
    const int wave = tid >> 5;
    const int colbase = wave * 16;
    const int r = lane & 15;
    const int hi = lane >> 4;                     // 0: K pair {0,1}, 1: K pair {2,3}
    const float* arow = &Alds[r * LDSA + hi * 2];
    const float2* bp = Wp + (size_t)hi * DFEAT + colbase + r;

    v8f acc = {};
    #pragma unroll 8
    for (int kk = 0; kk < 32; ++kk) {
        v2f a, b;
        a.x = arow[kk * 4 + 0];
        a.y = arow[kk * 4 + 1];
        float2 bv = bp[(size_t)kk * (2 * DFEAT)];
        b.x = bv.x; b.y = bv.y;
        acc = __builtin_amdgcn_wmma_f32_16x16x4_f32(false, a, false, b,
                                                    (short)0, acc, false, false);
    }

    // C/D layout: VGPR i -> row i (lanes 0-15) / row i+8 (lanes 16-31), col = lane&15
    const int rowoff = hi << 3;
    float* op = Y + (size_t)(rowbase + rowoff) * DFEAT + colbase + r;
    if (full) {
        #pragma unroll
        for (int i = 0; i < 8; ++i) op[(size_t)i * DFEAT] = acc[i];
    } else {
        #pragma unroll
        for (int i = 0; i < 8; ++i)
            if (rowbase + rowoff + i < n) op[(size_t)i * DFEAT] = acc[i];
    }
}

// ---------------- aggregation: init with self-loop + bias ----------------
__global__ __launch_bounds__(256) void agg_init_kernel(const float* __restrict__ H,
                                                       const float* __restrict__ dis,
                                                       const float* __restrict__ bias,
                                                       float* __restrict__ out, int n) {
    long long t = (long long)blockIdx.x * 256 + threadIdx.x;
    int i = (int)(t >> 5);
    int lane = (int)(t & 31);
    if (i >= n) return;
    float di = dis[i];
    float c = di * di;                      // norm for self loop
    float4 v = ((const float4*)(H + (size_t)i * DFEAT))[lane];
    float4 b = ((const float4*)bias)[lane];
    float4 o;
    o.x = v.x * c + b.x; o.y = v.y * c + b.y;
    o.z = v.z * c + b.z; o.w = v.w * c + b.w;
    ((float4*)(out + (size_t)i * DFEAT))[lane] = o;
}

// ---------------- aggregation: edge scatter-add (32 lanes per edge, L2-resident) ----------------
__global__ __launch_bounds__(256) void agg_edges_kernel(const float* __restrict__ H,
                                                        const long long* __restrict__ src,
                                                        const long long* __restrict__ dst,
                                                        const float* __restrict__ dis,
                                                        float* __restrict__ out, long long E) {
    long long t = (long long)blockIdx.x * 256 + threadIdx.x;
    long long e = t >> 5;
    int lane = (int)(t & 31);
    if (e >= E) return;
    int s = (int)src[e];
    int d = (int)dst[e];
    float coeff = dis[s] * dis[d];
    float4 v = ((const float4*)(H + (size_t)s * DFEAT))[lane];
    float* o = out + (size_t)d * DFEAT + lane * 4;
    atomicAdd(o + 0, v.x * coeff);
    atomicAdd(o + 1, v.y * coeff);
    atomicAdd(o + 2, v.z * coeff);
    atomicAdd(o + 3, v.w * coeff);
}

// ---------------- output layer: relu(H) @ Wout[128x40] + bout, then log_softmax ----------------
// 1 block = 16 rows, 3 waves (cols padded 40 -> 48, zero-padded packed weights).
__global__ __launch_bounds__(96) void out_kernel(const float* __restrict__ H,
                                                 const float2* __restrict__ WpO,
                                                 const float* __restrict__ bout,
                                                 float* __restrict__ out, int n) {
    __shared__ float Alds[16 * LDSA];
    __shared__ float Llds[16 * COLP];
    const int tid = threadIdx.x;
    const int rowbase = blockIdx.x * 16;

    for (int idx = tid; idx < 512; idx += 96) {
        int r = idx >> 5, c4 = idx & 31;
        int row = rowbase + r; if (row >= n) row = n - 1;
        float4 v = ((const float4*)(H + (size_t)row * DFEAT))[c4];
        v.x = fmaxf(v.x, 0.f); v.y = fmaxf(v.y, 0.f);
        v.z = fmaxf(v.z, 0.f); v.w = fmaxf(v.w, 0.f);
        float* dp = &Alds[r * LDSA + c4 * 4];
        dp[0] = v.x; dp[1] = v.y; dp[2] = v.z; dp[3] = v.w;
    }
    __syncthreads();

    const int lane = tid & 31;
    const int wave = tid >> 5;
    const int colbase = wave * 16;
    const int r = lane & 15;
    const int hi = lane >> 4;
    const int col = colbase + r;
    const float* arow = &Alds[r * LDSA + hi * 2];
    const float2* bp = WpO + (size_t)hi * COLP + col;

    v8f acc = {};
    #pragma unroll 8
    for (int kk = 0; kk < 32; ++kk) {
        v2f a, b;
        a.x = arow[kk * 4 + 0];
        a.y = arow[kk * 4 + 1];
        float2 bv = bp[(size_t)kk * (2 * COLP)];
        b.x = bv.x; b.y = bv.y;
        acc = __builtin_amdgcn_wmma_f32_16x16x4_f32(false, a, false, b,
                                                    (short)0, acc, false, false);
    }

    const int rowoff = hi << 3;
    if (col < DOUT) {
        float bb = bout[col];
        #pragma unroll
        for (int i = 0; i < 8; ++i) Llds[(rowoff + i) * COLP + col] = acc[i] + bb;
    }
    __syncthreads();

    if (tid < 16) {
        int row = rowbase + tid;
        if (row < n) {
            const float* l = &Llds[tid * COLP];
            float m = -3.402823466e38f;
            for (int c = 0; c < DOUT; ++c) m = fmaxf(m, l[c]);
            float s = 0.f;
            for (int c = 0; c < DOUT; ++c) s += expf(l[c] - m);
            float ls = logf(s);
            float* op = out + (size_t)row * DOUT;
            for (int c = 0; c < DOUT; ++c) op[c] = l[c] - m - ls;
        }
    }
}

extern "C" void kernel_launch(void* const* d_in, const int* in_sizes, int n_in,
                              void* d_out, int out_size, void* d_ws, size_t ws_size,
                              hipStream_t stream) {
    const float*     x    = (const float*)d_in[0];
    const long long* ei   = (const long long*)d_in[1];
    // d_in[2] = batch (unused)
    const float*     W1   = (const float*)d_in[3];
    const float*     b1   = (const float*)d_in[4];
    const float*     W2   = (const float*)d_in[5];
    const float*     b2   = (const float*)d_in[6];
    const float*     Wout = (const float*)d_in[7];
    const float*     bout = (const float*)d_in[8];

    const int       N = in_sizes[0] / DFEAT;
    const long long E = (long long)in_sizes[1] / 2;
    const long long* src = ei;
    const long long* dst = ei + E;

    char* wsp = (char*)d_ws;
    size_t off = 0;
    float* dis  = (float*)(wsp + off); off += (((size_t)N * 4) + 255) & ~(size_t)255;
    float* bufA = (float*)(wsp + off); off += (((size_t)N * DFEAT * 4) + 255) & ~(size_t)255;
    float* bufB = (float*)(wsp + off); off += (((size_t)N * DFEAT * 4) + 255) & ~(size_t)255;
    float* Wp   = (float*)(wsp + off); off += (size_t)DFEAT * DFEAT * 4;       // 64 KB
    float* WpO  = (float*)(wsp + off); off += (size_t)DFEAT * COLP * 4;        // 24 KB

    const int nb256   = (N + 255) / 256;
    const int eb      = (int)((E + 255) / 256);
    const int gb      = (N + 15) / 16;
    const int aggInit = (int)((((long long)N * 32) + 255) / 256);
    const int aggEdge = (int)(((E * 32) + 255) / 256);
    const int pk128   = (DFEAT * DFEAT + 255) / 256;
    const int pkOut   = (DFEAT * COLP + 255) / 256;

    // normalization: deg -> dis (in place)
    deg_init_kernel<<<nb256, 256, 0, stream>>>(dis, N);
    deg_edges_kernel<<<eb, 256, 0, stream>>>(dst, dis, E);
    dis_finish_kernel<<<nb256, 256, 0, stream>>>(dis, N);

    // layer 1
    pack_w128_kernel<<<pk128, 256, 0, stream>>>(W1, Wp);
    gemm128_kernel<<<gb, 256, 0, stream>>>(x, (const float2*)Wp, bufA, N, 0);
    agg_init_kernel<<<aggInit, 256, 0, stream>>>(bufA, dis, b1, bufB, N);
    agg_edges_kernel<<<aggEdge, 256, 0, stream>>>(bufA, src, dst, dis, bufB, E);

    // layer 2 (relu fused into A-load)
    pack_w128_kernel<<<pk128, 256, 0, stream>>>(W2, Wp);
    gemm128_kernel<<<gb, 256, 0, stream>>>(bufB, (const float2*)Wp, bufA, N, 1);
    agg_init_kernel<<<aggInit, 256, 0, stream>>>(bufA, dis, b2, bufB, N);
    agg_edges_kernel<<<aggEdge, 256, 0, stream>>>(bufA, src, dst, dis, bufB, E);

    // output layer + log_softmax
    pack_wout_kernel<<<pkOut, 256, 0, stream>>>(Wout, WpO);
    out_kernel<<<gb, 96, 0, stream>>>(bufB, (const float2*)WpO, bout, (float*)d_out, N);
}